// SingleModel_73675868995972
// MI455X (gfx1250) — compile-verified
//
#include <hip/hip_runtime.h>
#include <math.h>

// Problem dimensions (fixed by the reference)
#define NI   100000
#define NFN  2000
#define NE   800000
#define NEF  200000
#define NH   64
#define EH   8
#define KA   72    // NH + EH
#define NL   11

typedef float v2f __attribute__((ext_vector_type(2)));
typedef float v8f __attribute__((ext_vector_type(8)));

__device__ __forceinline__ float elu1(float x) {
    return x > 0.0f ? x : (expf(x) - 1.0f);
}

// ---------------------------------------------------------------------------
// Node encoder: Y[n,64] = elu(X[n,16] @ W[16,64] + b)
// ---------------------------------------------------------------------------
__global__ void enc_nodes_k(const float* __restrict__ X, const float* __restrict__ W,
                            const float* __restrict__ b, float* __restrict__ Y, int n) {
    int t = blockIdx.x * blockDim.x + threadIdx.x;
    if (t >= n * 64) return;
    int i = t >> 6, j = t & 63;
    float acc = b[j];
    const float* x = X + i * 16;
#pragma unroll
    for (int k = 0; k < 16; ++k) acc = fmaf(x[k], W[k * 64 + j], acc);
    Y[t] = elu1(acc);
}

// ---------------------------------------------------------------------------
// Edge encoder + scatter-sum (prev & succ directions) + degree counts.
// e_prev = elu(ef @ Wp + bp) scattered onto prev_dst,
// e_succ = elu(ef @ Ws + bs) scattered onto prev_src.
// ---------------------------------------------------------------------------
__global__ void edge_prevsucc_k(const float* __restrict__ ef,
                                const int* __restrict__ src, const int* __restrict__ dst,
                                const float* __restrict__ Wp, const float* __restrict__ bp,
                                const float* __restrict__ Ws, const float* __restrict__ bs,
                                float* AeP, float* AeS, float* degP, float* degS) {
    int e = blockIdx.x * blockDim.x + threadIdx.x;
    if (e >= NE) return;
    float x[8];
#pragma unroll
    for (int k = 0; k < 8; ++k) x[k] = ef[e * 8 + k];
    int s = src[e], d = dst[e];
#pragma unroll
    for (int j = 0; j < 8; ++j) {
        float ap = bp[j], as2 = bs[j];
#pragma unroll
        for (int k = 0; k < 8; ++k) {
            ap  = fmaf(x[k], Wp[k * 8 + j], ap);
            as2 = fmaf(x[k], Ws[k * 8 + j], as2);
        }
        atomicAdd(&AeP[d * 8 + j], elu1(ap));
        atomicAdd(&AeS[s * 8 + j], elu1(as2));
    }
    atomicAdd(&degP[d], 1.0f);
    atomicAdd(&degS[s], 1.0f);
}

__global__ void edge_tf_k(const float* __restrict__ ef,
                          const int* __restrict__ dst,
                          const float* __restrict__ Wf, const float* __restrict__ bf,
                          float* AeT, float* degT) {
    int e = blockIdx.x * blockDim.x + threadIdx.x;
    if (e >= NEF) return;
    float x[8];
#pragma unroll
    for (int k = 0; k < 8; ++k) x[k] = ef[e * 8 + k];
    int d = dst[e];
#pragma unroll
    for (int j = 0; j < 8; ++j) {
        float a = bf[j];
#pragma unroll
        for (int k = 0; k < 8; ++k) a = fmaf(x[k], Wf[k * 8 + j], a);
        atomicAdd(&AeT[d * 8 + j], elu1(a));
    }
    atomicAdd(&degT[d], 1.0f);
}

// deg -> 1/max(deg,1) in place; Ae *= invdeg (mean). Layer-invariant, done once.
__global__ void normalize_k(float* Ae, float* deg, int n) {
    int i = blockIdx.x * blockDim.x + threadIdx.x;
    if (i >= n) return;
    float inv = 1.0f / fmaxf(deg[i], 1.0f);
    deg[i] = inv;
#pragma unroll
    for (int j = 0; j < 8; ++j) Ae[i * 8 + j] *= inv;
}

// ---------------------------------------------------------------------------
// Per-layer scatter of node features (both directions in one pass).
// Thread t -> edge t/64, feature t%64.
// ---------------------------------------------------------------------------
__global__ void scatter_prevsucc_k(const float* __restrict__ inst,
                                   const int* __restrict__ src, const int* __restrict__ dst,
                                   float* AccP, float* AccS) {
    int t = blockIdx.x * blockDim.x + threadIdx.x;
    int e = t >> 6;
    if (e >= NE) return;
    int j = t & 63;
    int s = src[e], d = dst[e];
    atomicAdd(&AccP[d * 64 + j], inst[s * 64 + j]);
    atomicAdd(&AccS[s * 64 + j], inst[d * 64 + j]);
}

__global__ void scatter_tf_k(const float* __restrict__ inst,
                             const int* __restrict__ src, const int* __restrict__ dst,
                             float* AccT) {
    int t = blockIdx.x * blockDim.x + threadIdx.x;
    int e = t >> 6;
    if (e >= NEF) return;
    int j = t & 63;
    atomicAdd(&AccT[dst[e] * 64 + j], inst[src[e] * 64 + j]);
}

// ---------------------------------------------------------------------------
// WMMA layer update for instruction nodes:
//   o_p = [AccP/deg | AeP] @ Wp + bp ; o_s = [AccS/deg | AeS] @ Ws + bs
//   inst = elu(inst + 0.5*(o_p + o_s))
// Block: 128 threads = 4 waves; block owns a 16-node row tile, wave w owns
// columns [16w,16w+16). K = 72, reduced with 18 V_WMMA_F32_16X16X4_F32 steps
// per GEMM (36 WMMAs per wave).
// ---------------------------------------------------------------------------
__global__ void __launch_bounds__(128)
layer_inst_k(const float* __restrict__ AccP, const float* __restrict__ AccS,
             const float* __restrict__ invdP, const float* __restrict__ invdS,
             const float* __restrict__ AeP, const float* __restrict__ AeS,
             const float* __restrict__ Wp, const float* __restrict__ bp,
             const float* __restrict__ Ws, const float* __restrict__ bs,
             float* __restrict__ inst) {
    __shared__ float As[2][16][80];  // 72 padded to 80 (bank-conflict avoidance)
    int row0 = blockIdx.x * 16;
    for (int idx = threadIdx.x; idx < 16 * KA; idx += 128) {
        int r = idx / KA, k = idx % KA;
        int node = row0 + r;
        float vP, vS;
        if (k < 64) {
            vP = AccP[node * 64 + k] * invdP[node];
            vS = AccS[node * 64 + k] * invdS[node];
        } else {
            vP = AeP[node * 8 + (k - 64)];
            vS = AeS[node * 8 + (k - 64)];
        }
        As[0][r][k] = vP;
        As[1][r][k] = vS;
    }
    __syncthreads();

    int lane = threadIdx.x & 31;
    int wv   = threadIdx.x >> 5;
    int m    = lane & 15;
    int kg   = (lane >> 4) << 1;   // 0 or 2 (K sub-offset per lane group)
    int col  = wv * 16 + m;

    v8f cp = {};
    v8f cs = {};
#pragma unroll
    for (int kb = 0; kb < KA; kb += 4) {
        v2f aP, aS, bP, bS;
        aP[0] = As[0][m][kb + kg];     aP[1] = As[0][m][kb + kg + 1];
        aS[0] = As[1][m][kb + kg];     aS[1] = As[1][m][kb + kg + 1];
        const float* w0 = Wp + (kb + kg) * 64 + col;
        bP[0] = w0[0];  bP[1] = w0[64];
        const float* w1 = Ws + (kb + kg) * 64 + col;
        bS[0] = w1[0];  bS[1] = w1[64];
        cp = __builtin_amdgcn_wmma_f32_16x16x4_f32(false, aP, false, bP, (short)0, cp, false, false);
        cs = __builtin_amdgcn_wmma_f32_16x16x4_f32(false, aS, false, bS, (short)0, cs, false, false);
    }

    float bsum = 0.5f * (bp[col] + bs[col]);
    int g8 = (lane >> 4) * 8;  // D layout: lanes 16-31 hold rows M+8
#pragma unroll
    for (int v = 0; v < 8; ++v) {
        int node = row0 + v + g8;
        float* p = inst + node * 64 + col;
        *p = elu1(*p + 0.5f * (cp[v] + cs[v]) + bsum);
    }
}

// Same structure, single GEMM, residual-updates the final-node states.
__global__ void __launch_bounds__(128)
layer_fin_k(const float* __restrict__ AccT, const float* __restrict__ invdT,
            const float* __restrict__ AeT,
            const float* __restrict__ Wf, const float* __restrict__ bf,
            float* __restrict__ fin) {
    __shared__ float As[16][80];
    int row0 = blockIdx.x * 16;
    for (int idx = threadIdx.x; idx < 16 * KA; idx += 128) {
        int r = idx / KA, k = idx % KA;
        int node = row0 + r;
        As[r][k] = (k < 64) ? AccT[node * 64 + k] * invdT[node]
                            : AeT[node * 8 + (k - 64)];
    }
    __syncthreads();

    int lane = threadIdx.x & 31;
    int wv   = threadIdx.x >> 5;
    int m    = lane & 15;
    int kg   = (lane >> 4) << 1;
    int col  = wv * 16 + m;

    v8f c = {};
#pragma unroll
    for (int kb = 0; kb < KA; kb += 4) {
        v2f a, bv;
        a[0] = As[m][kb + kg];  a[1] = As[m][kb + kg + 1];
        const float* w0 = Wf + (kb + kg) * 64 + col;
        bv[0] = w0[0];  bv[1] = w0[64];
        c = __builtin_amdgcn_wmma_f32_16x16x4_f32(false, a, false, bv, (short)0, c, false, false);
    }

    float bb = bf[col];
    int g8 = (lane >> 4) * 8;
#pragma unroll
    for (int v = 0; v < 8; ++v) {
        int node = row0 + v + g8;
        float* p = fin + node * 64 + col;
        *p = elu1(*p + c[v] + bb);
    }
}

// Readout dense: Y[n,64] = elu(X[n,64] @ W[64,64] + b), n multiple of 16.
__global__ void __launch_bounds__(128)
dense64_k(const float* __restrict__ X, const float* __restrict__ W,
          const float* __restrict__ b, float* __restrict__ Y) {
    __shared__ float As[16][68];
    int row0 = blockIdx.x * 16;
    for (int idx = threadIdx.x; idx < 16 * 64; idx += 128) {
        int r = idx >> 6, k = idx & 63;
        As[r][k] = X[(row0 + r) * 64 + k];
    }
    __syncthreads();

    int lane = threadIdx.x & 31;
    int wv   = threadIdx.x >> 5;
    int m    = lane & 15;
    int kg   = (lane >> 4) << 1;
    int col  = wv * 16 + m;

    v8f c = {};
#pragma unroll
    for (int kb = 0; kb < 64; kb += 4) {
        v2f a, bv;
        a[0] = As[m][kb + kg];  a[1] = As[m][kb + kg + 1];
        const float* w0 = W + (kb + kg) * 64 + col;
        bv[0] = w0[0];  bv[1] = w0[64];
        c = __builtin_amdgcn_wmma_f32_16x16x4_f32(false, a, false, bv, (short)0, c, false, false);
    }

    float bb = b[col];
    int g8 = (lane >> 4) * 8;
#pragma unroll
    for (int v = 0; v < 8; ++v)
        Y[(row0 + v + g8) * 64 + col] = elu1(c[v] + bb);
}

// Final projection to scalar per node: out[i] = X[i,:64] . w + b
__global__ void readout_k(const float* __restrict__ X, const float* __restrict__ w,
                          const float* __restrict__ b, float* __restrict__ out) {
    int i = blockIdx.x * blockDim.x + threadIdx.x;
    if (i >= NFN) return;
    float acc = b[0];
#pragma unroll
    for (int k = 0; k < 64; ++k) acc = fmaf(X[i * 64 + k], w[k], acc);
    out[i] = acc;
}

// ---------------------------------------------------------------------------
// Host orchestration
// ---------------------------------------------------------------------------
static inline size_t al256(size_t x) { return (x + 255) & ~(size_t)255; }

extern "C" void kernel_launch(void* const* d_in, const int* in_sizes, int n_in,
                              void* d_out, int out_size, void* d_ws, size_t ws_size,
                              hipStream_t stream) {
    (void)in_sizes; (void)n_in; (void)out_size; (void)ws_size;

    const float* inst_x = (const float*)d_in[0];
    const float* fin_x  = (const float*)d_in[1];
    const float* e_ins  = (const float*)d_in[2];
    const float* e_tf   = (const float*)d_in[3];
    const int*   prev_src = (const int*)d_in[4];
    const int*   prev_dst = (const int*)d_in[5];
    const int*   tf_src   = (const int*)d_in[6];
    const int*   tf_dst   = (const int*)d_in[7];
    const float* W_it = (const float*)d_in[8];   const float* b_it = (const float*)d_in[9];
    const float* W_ft = (const float*)d_in[10];  const float* b_ft = (const float*)d_in[11];
    const float* W_ep = (const float*)d_in[12];  const float* b_ep = (const float*)d_in[13];
    const float* W_es = (const float*)d_in[14];  const float* b_es = (const float*)d_in[15];
    const float* W_ef = (const float*)d_in[16];  const float* b_ef = (const float*)d_in[17];
    const float* Wg_prev = (const float*)d_in[18]; const float* bg_prev = (const float*)d_in[19];
    const float* Wg_succ = (const float*)d_in[20]; const float* bg_succ = (const float*)d_in[21];
    const float* Wg_tf   = (const float*)d_in[22]; const float* bg_tf   = (const float*)d_in[23];
    const float* Wr1 = (const float*)d_in[24]; const float* br1 = (const float*)d_in[25];
    const float* Wr2 = (const float*)d_in[26]; const float* br2 = (const float*)d_in[27];
    const float* Wr3 = (const float*)d_in[28]; const float* br3 = (const float*)d_in[29];
    float* out = (float*)d_out;

    // Workspace bump allocation (~86 MB)
    char* ws = (char*)d_ws;
    size_t off = 0;
    auto alloc = [&](size_t bytes) -> float* {
        float* p = (float*)(ws + off);
        off += al256(bytes);
        return p;
    };
    float* inst = alloc((size_t)NI * 64 * 4);
    float* fin  = alloc((size_t)NFN * 64 * 4);
    float* fin2 = alloc((size_t)NFN * 64 * 4);
    float* AeP  = alloc((size_t)NI * 8 * 4);
    float* AeS  = alloc((size_t)NI * 8 * 4);
    float* AeT  = alloc((size_t)NFN * 8 * 4);
    float* degP = alloc((size_t)NI * 4);
    float* degS = alloc((size_t)NI * 4);
    float* degT = alloc((size_t)NFN * 4);
    float* AccP = alloc((size_t)NI * 64 * 4);
    float* AccS = alloc((size_t)NI * 64 * 4);
    float* AccT = alloc((size_t)NFN * 64 * 4);

    // ---- Encoders ----
    enc_nodes_k<<<(NI * 64 + 255) / 256, 256, 0, stream>>>(inst_x, W_it, b_it, inst, NI);
    enc_nodes_k<<<(NFN * 64 + 255) / 256, 256, 0, stream>>>(fin_x, W_ft, b_ft, fin, NFN);

    // ---- Layer-invariant edge aggregates + degrees ----
    hipMemsetAsync(AeP, 0, (size_t)NI * 8 * 4, stream);
    hipMemsetAsync(AeS, 0, (size_t)NI * 8 * 4, stream);
    hipMemsetAsync(AeT, 0, (size_t)NFN * 8 * 4, stream);
    hipMemsetAsync(degP, 0, (size_t)NI * 4, stream);
    hipMemsetAsync(degS, 0, (size_t)NI * 4, stream);
    hipMemsetAsync(degT, 0, (size_t)NFN * 4, stream);
    edge_prevsucc_k<<<(NE + 255) / 256, 256, 0, stream>>>(
        e_ins, prev_src, prev_dst, W_ep, b_ep, W_es, b_es, AeP, AeS, degP, degS);
    edge_tf_k<<<(NEF + 255) / 256, 256, 0, stream>>>(e_tf, tf_dst, W_ef, b_ef, AeT, degT);
    normalize_k<<<(NI + 255) / 256, 256, 0, stream>>>(AeP, degP, NI);
    normalize_k<<<(NI + 255) / 256, 256, 0, stream>>>(AeS, degS, NI);
    normalize_k<<<(NFN + 255) / 256, 256, 0, stream>>>(AeT, degT, NFN);

    // ---- 11 message-passing layers ----
    for (int l = 0; l < NL; ++l) {
        hipMemsetAsync(AccP, 0, (size_t)NI * 64 * 4, stream);
        hipMemsetAsync(AccS, 0, (size_t)NI * 64 * 4, stream);
        hipMemsetAsync(AccT, 0, (size_t)NFN * 64 * 4, stream);
        scatter_prevsucc_k<<<(NE * 64) / 256, 256, 0, stream>>>(inst, prev_src, prev_dst, AccP, AccS);
        scatter_tf_k<<<(NEF * 64) / 256, 256, 0, stream>>>(inst, tf_src, tf_dst, AccT);
        layer_fin_k<<<NFN / 16, 128, 0, stream>>>(
            AccT, degT, AeT, Wg_tf + (size_t)l * KA * 64, bg_tf + (size_t)l * 64, fin);
        layer_inst_k<<<NI / 16, 128, 0, stream>>>(
            AccP, AccS, degP, degS, AeP, AeS,
            Wg_prev + (size_t)l * KA * 64, bg_prev + (size_t)l * 64,
            Wg_succ + (size_t)l * KA * 64, bg_succ + (size_t)l * 64, inst);
    }

    // ---- Readout MLP ----
    dense64_k<<<NFN / 16, 128, 0, stream>>>(fin, Wr1, br1, fin2);
    dense64_k<<<NFN / 16, 128, 0, stream>>>(fin2, Wr2, br2, fin);
    readout_k<<<(NFN + 255) / 256, 256, 0, stream>>>(fin, Wr3, br3, out);
}